// RandLANet_59201829208480
// MI455X (gfx1250) — compile-verified
//
#include <hip/hip_runtime.h>

typedef float v2f __attribute__((ext_vector_type(2)));
typedef float v8f __attribute__((ext_vector_type(8)));

constexpr int BB   = 2;
constexpr int NN   = 32768;
constexpr int KNB  = 16;      // neighbors
constexpr float EPSF = 1e-6f;
constexpr int WAVES = 8;      // waves per block (256 threads)

// ---------------- workspace layout (in floats) ----------------
constexpr size_t X1_OFF  = 0;                            // (B,N,32)
constexpr size_t X2_OFF  = X1_OFF + (size_t)BB*NN*32;    // (B,N,32)
constexpr size_t SC_OFF  = X2_OFF + (size_t)BB*NN*32;    // (B,128,N)
constexpr size_t P1_OFF  = SC_OFF + (size_t)BB*128*NN;   // (B,N,32)
constexpr size_t P2_OFF  = P1_OFF + (size_t)BB*NN*32;    // (B,N,64)
constexpr size_t ST_OFF  = P2_OFF + (size_t)BB*NN*64;    // 5 * (B,16,2)

__device__ __forceinline__ v8f v8z() {
  v8f z = {0.f,0.f,0.f,0.f,0.f,0.f,0.f,0.f};
  return z;
}

__device__ __forceinline__ v8f wmma4(v2f a, v2f b, v8f c) {
  // D = A(16x4, f32) x B(4x16, f32) + C(16x16, f32)
  return __builtin_amdgcn_wmma_f32_16x16x4_f32(false, a, false, b, (short)0, c,
                                               false, false);
}

// A fragment: row-major weight W[M][kdim]; lane -> m = mbase + lane%16,
// vgpr v -> k = kk + v + 2*(lane/16). Zero-pad past kdim.
__device__ __forceinline__ v2f afrag(const float* __restrict__ W, int ld,
                                     int mbase, int kk, int kdim, int lane) {
  int m  = mbase + (lane & 15);
  int k0 = kk + 2 * (lane >> 4);
  v2f a;
  a.x = (k0     < kdim) ? W[m * ld + k0]     : 0.f;
  a.y = (k0 + 1 < kdim) ? W[m * ld + k0 + 1] : 0.f;
  return a;
}

__device__ __forceinline__ float rsum16(float v) {
  v += __shfl_xor(v, 1, 16);
  v += __shfl_xor(v, 2, 16);
  v += __shfl_xor(v, 4, 16);
  v += __shfl_xor(v, 8, 16);
  return v;
}

// concat channel selector: [ctr(3), nbr(3), ctr-nbr(3), dist(1), pad...]
__device__ __forceinline__ float xsel(int k, float cx, float cy, float cz,
                                      float nx, float ny, float nz, float dj) {
  switch (k) {
    case 0: return cx; case 1: return cy; case 2: return cz;
    case 3: return nx; case 4: return ny; case 5: return nz;
    case 6: return cx - nx; case 7: return cy - ny; case 8: return cz - nz;
    case 9: return dj; default: return 0.f;
  }
}

// acc[0..1] += W(32x10) * X(10x16) for one point's 16 neighbors (pre-bias).
__device__ __forceinline__ void enc_gemm(const float* __restrict__ coords,
                                         const int* __restrict__ idx,
                                         const float* __restrict__ dist,
                                         const float* __restrict__ w,
                                         int b, int n, int lane, v8f acc[2]) {
  int h = lane >> 4, j = lane & 15;
  int pb = b * NN + n;
  float cx = coords[pb * 3 + 0];
  float cy = coords[pb * 3 + 1];
  float cz = coords[pb * 3 + 2];
  int ij = idx[pb * KNB + j];
  int nb = b * NN + ij;
  float nx = coords[nb * 3 + 0];
  float ny = coords[nb * 3 + 1];
  float nz = coords[nb * 3 + 2];
  float dj = dist[pb * KNB + j];
#pragma unroll
  for (int kk = 0; kk < 12; kk += 4) {
    v2f bf;
    if (h == 0) {
      bf.x = xsel(kk + 0, cx, cy, cz, nx, ny, nz, dj);
      bf.y = xsel(kk + 1, cx, cy, cz, nx, ny, nz, dj);
    } else {
      bf.x = xsel(kk + 2, cx, cy, cz, nx, ny, nz, dj);
      bf.y = xsel(kk + 3, cx, cy, cz, nx, ny, nz, dj);
    }
#pragma unroll
    for (int t = 0; t < 2; ++t)
      acc[t] = wmma4(afrag(w, 10, t * 16, kk, 10, lane), bf, acc[t]);
  }
}

// ------------------------------------------------------------------
// stage0: x1 = leaky(mlp1(features),0.2) (B,N,32); sc_raw (B,128,N) + sc GN stats
// ------------------------------------------------------------------
__global__ __launch_bounds__(256) void k_stage0(
    const float* __restrict__ feat,      // (B,72,N)
    const float* __restrict__ w_mlp1, const float* __restrict__ b_mlp1,
    const float* __restrict__ w_sc,   const float* __restrict__ b_sc,
    float* __restrict__ x1, float* __restrict__ scraw,
    float* __restrict__ st_sc) {
  __shared__ float s_sum[16], s_sq[16];
  int tid = threadIdx.x;
  if (tid < 16) { s_sum[tid] = 0.f; s_sq[tid] = 0.f; }
  __syncthreads();
  int lane = tid & 31, wv = tid >> 5;
  int gwave = blockIdx.x * WAVES + wv;
  int b = gwave / (NN / 16);
  int nbase = (gwave % (NN / 16)) * 16;
  int h = lane >> 4, col = lane & 15;
  int n = nbase + col;

  v8f am[2]; am[0] = v8z(); am[1] = v8z();
  v8f as[8];
#pragma unroll
  for (int t = 0; t < 8; ++t) as[t] = v8z();

  for (int kk = 0; kk < 72; kk += 4) {
    int k0 = kk + 2 * h;
    v2f bf;
    bf.x = feat[(b * 72 + k0) * NN + n];
    bf.y = feat[(b * 72 + k0 + 1) * NN + n];
#pragma unroll
    for (int t = 0; t < 2; ++t)
      am[t] = wmma4(afrag(w_mlp1, 72, t * 16, kk, 72, lane), bf, am[t]);
#pragma unroll
    for (int t = 0; t < 8; ++t)
      as[t] = wmma4(afrag(w_sc, 72, t * 16, kk, 72, lane), bf, as[t]);
  }
  // x1 = leaky_relu(conv + bias, 0.2), stored (B,N,32)
#pragma unroll
  for (int t = 0; t < 2; ++t)
#pragma unroll
    for (int r = 0; r < 8; ++r) {
      int c = t * 16 + r + 8 * h;
      float v = am[t][r] + b_mlp1[c];
      v = v > 0.f ? v : 0.2f * v;
      x1[(b * NN + n) * 32 + c] = v;
    }
  // shortcut pre-GN, stored (B,128,N); accumulate GN stats (8 ch / group)
#pragma unroll
  for (int t = 0; t < 8; ++t) {
    float ls = 0.f, lq = 0.f;
#pragma unroll
    for (int r = 0; r < 8; ++r) {
      int o = t * 16 + r + 8 * h;
      float v = as[t][r] + b_sc[o];
      scraw[(b * 128 + o) * NN + n] = v;
      ls += v; lq += v * v;
    }
    ls = rsum16(ls); lq = rsum16(lq);
    if (col == 0) {
      int g = 2 * t + h;   // 8 consecutive channels per half-wave = 1 group
      atomicAdd(&s_sum[g], ls);
      atomicAdd(&s_sq[g], lq);
    }
  }
  __syncthreads();
  if (tid < 16) {
    atomicAdd(&st_sc[(b * 16 + tid) * 2 + 0], s_sum[tid]);
    atomicAdd(&st_sc[(b * 16 + tid) * 2 + 1], s_sq[tid]);
  }
}

// ------------------------------------------------------------------
// lse stats: sum / sumsq of conv1x1(concat)+bias per (b, group)
// ------------------------------------------------------------------
__global__ __launch_bounds__(256) void k_lse_stats(
    const float* __restrict__ coords, const int* __restrict__ idx,
    const float* __restrict__ dist,
    const float* __restrict__ w, const float* __restrict__ bias,
    float* __restrict__ st) {
  __shared__ float s_sum[16], s_sq[16];
  int tid = threadIdx.x;
  if (tid < 16) { s_sum[tid] = 0.f; s_sq[tid] = 0.f; }
  __syncthreads();
  int lane = tid & 31, wv = tid >> 5;
  int gp = blockIdx.x * WAVES + wv;   // one point per wave
  int b = gp / NN, n = gp % NN;
  int h = lane >> 4, j = lane & 15;

  v8f acc[2]; acc[0] = v8z(); acc[1] = v8z();
  enc_gemm(coords, idx, dist, w, b, n, lane, acc);

#pragma unroll
  for (int t = 0; t < 2; ++t)
#pragma unroll
    for (int r = 0; r < 8; r += 2) {
      int m = t * 16 + r + 8 * h;          // even -> group m>>1
      float v0 = acc[t][r]     + bias[m];
      float v1 = acc[t][r + 1] + bias[m + 1];
      float ls = rsum16(v0 + v1);
      float lq = rsum16(v0 * v0 + v1 * v1);
      if (j == 0) {
        atomicAdd(&s_sum[m >> 1], ls);
        atomicAdd(&s_sq[m >> 1], lq);
      }
    }
  __syncthreads();
  if (tid < 16) {
    atomicAdd(&st[(b * 16 + tid) * 2 + 0], s_sum[tid]);
    atomicAdd(&st[(b * 16 + tid) * 2 + 1], s_sq[tid]);
  }
}

// ------------------------------------------------------------------
// lse apply + attentive pool: recompute enc, GN+relu, mean over K,
// gather-mean neighbor feats, pool conv (WMMA) -> praw + pool GN stats.
// ------------------------------------------------------------------
template <int PO>
__global__ __launch_bounds__(256) void k_lse_apply_pool(
    const float* __restrict__ coords, const int* __restrict__ idx,
    const float* __restrict__ dist,
    const float* __restrict__ w_lse, const float* __restrict__ b_lse,
    const float* __restrict__ g_lse, const float* __restrict__ bt_lse,
    const float* __restrict__ st_lse,
    const float* __restrict__ xprev,              // (B,N,32)
    const float* __restrict__ w_pool, const float* __restrict__ b_pool,
    float* __restrict__ praw,                     // (B,N,PO)
    float* __restrict__ st_pool) {
  __shared__ float s_sum[16], s_sq[16];
  __shared__ float p_sh[WAVES][16][68];           // [wave][pt][64ch] padded
  int tid = threadIdx.x;
  if (tid < 16) { s_sum[tid] = 0.f; s_sq[tid] = 0.f; }
  __syncthreads();
  int lane = tid & 31, wv = tid >> 5;
  int gwave = blockIdx.x * WAVES + wv;
  int b = gwave / (NN / 16);
  int nbase = (gwave % (NN / 16)) * 16;
  int h = lane >> 4, col = lane & 15;
  float (*psh)[68] = p_sh[wv];

  // hoist GN scale/shift per held enc channel (bias folded in)
  const float inv_lse = 1.f / (2.f * NN * KNB);
  float esc[2][8], esh[2][8];
#pragma unroll
  for (int t = 0; t < 2; ++t)
#pragma unroll
    for (int r = 0; r < 8; ++r) {
      int m = t * 16 + r + 8 * h;
      int g = m >> 1;
      float s  = st_lse[(b * 16 + g) * 2 + 0];
      float q  = st_lse[(b * 16 + g) * 2 + 1];
      float mu = s * inv_lse;
      float var = q * inv_lse - mu * mu;
      float rs = rsqrtf(var + EPSF);
      float gm = g_lse[m] * rs;
      esc[t][r] = gm;
      esh[t][r] = bt_lse[m] + (b_lse[m] - mu) * gm;
    }

  for (int pt = 0; pt < 16; ++pt) {
    int n = nbase + pt;
    v8f acc[2]; acc[0] = v8z(); acc[1] = v8z();
    enc_gemm(coords, idx, dist, w_lse, b, n, lane, acc);
#pragma unroll
    for (int t = 0; t < 2; ++t)
#pragma unroll
      for (int r = 0; r < 8; ++r) {
        float v = fmaxf(acc[t][r] * esc[t][r] + esh[t][r], 0.f);
        float rsm = rsum16(v);
        if (col == 0) psh[pt][t * 16 + r + 8 * h] = rsm * (1.f / 16.f);
      }
    // neighbor-feature mean: lane == channel (32 lanes, coalesced per j)
    int pb = b * NN + n;
    float fv = 0.f;
#pragma unroll
    for (int jj = 0; jj < 16; ++jj) {
      int ij = idx[pb * KNB + jj];
      fv += xprev[(b * NN + ij) * 32 + lane];
    }
    psh[pt][32 + lane] = fv * (1.f / 16.f);
  }
  __syncthreads();

  constexpr int PT = PO / 16;
  v8f ap[PT];
#pragma unroll
  for (int t = 0; t < PT; ++t) ap[t] = v8z();
#pragma unroll
  for (int kk = 0; kk < 64; kk += 4) {
    int k0 = kk + 2 * h;
    v2f bf;
    bf.x = psh[col][k0];
    bf.y = psh[col][k0 + 1];
#pragma unroll
    for (int t = 0; t < PT; ++t)
      ap[t] = wmma4(afrag(w_pool, 64, t * 16, kk, 64, lane), bf, ap[t]);
  }
  constexpr int CPG = PO / 16;   // GN channels per group
#pragma unroll
  for (int t = 0; t < PT; ++t)
#pragma unroll
    for (int r = 0; r < 8; ++r) {
      int o = t * 16 + r + 8 * h;
      float v = ap[t][r] + b_pool[o];
      praw[(b * NN + nbase + col) * PO + o] = v;
      float ls = rsum16(v);
      float lq = rsum16(v * v);
      if (col == 0) {
        atomicAdd(&s_sum[o / CPG], ls);
        atomicAdd(&s_sq[o / CPG], lq);
      }
    }
  __syncthreads();
  if (tid < 16) {
    atomicAdd(&st_pool[(b * 16 + tid) * 2 + 0], s_sum[tid]);
    atomicAdd(&st_pool[(b * 16 + tid) * 2 + 1], s_sq[tid]);
  }
}

// ------------------------------------------------------------------
// elementwise GN + ReLU (pool1 -> x2)
// ------------------------------------------------------------------
__global__ __launch_bounds__(256) void k_gn_relu(
    const float* __restrict__ in, float* __restrict__ out,
    const float* __restrict__ st, const float* __restrict__ gamma,
    const float* __restrict__ beta, float inv_cnt, int C, int total) {
  int i = blockIdx.x * blockDim.x + threadIdx.x;
  if (i >= total) return;
  int c = i % C;
  int bn = i / C;
  int b = bn / NN;
  int g = c / (C / 16);
  float s  = st[(b * 16 + g) * 2 + 0];
  float q  = st[(b * 16 + g) * 2 + 1];
  float mu = s * inv_cnt;
  float var = q * inv_cnt - mu * mu;
  float rs = rsqrtf(var + EPSF);
  float v = (in[i] - mu) * rs * gamma[c] + beta[c];
  out[i] = fmaxf(v, 0.f);
}

// ------------------------------------------------------------------
// final: x3 = GN+relu(pool2_raw); main = mlp2(x3); sc = GN(sc_raw);
// out = leaky(main + sc, 0.01)   out layout (B,128,N)
// ------------------------------------------------------------------
__global__ __launch_bounds__(256) void k_final(
    const float* __restrict__ p2raw,              // (B,N,64)
    const float* __restrict__ st_p2, const float* __restrict__ g_p2,
    const float* __restrict__ bt_p2,
    const float* __restrict__ w_mlp2, const float* __restrict__ b_mlp2,
    const float* __restrict__ scraw,              // (B,128,N)
    const float* __restrict__ st_sc, const float* __restrict__ g_sc,
    const float* __restrict__ bt_sc,
    float* __restrict__ out) {
  __shared__ float p_sh[WAVES][16][68];
  int tid = threadIdx.x;
  int lane = tid & 31, wv = tid >> 5;
  int gwave = blockIdx.x * WAVES + wv;
  int b = gwave / (NN / 16);
  int nbase = (gwave % (NN / 16)) * 16;
  int h = lane >> 4, col = lane & 15;
  float (*psh)[68] = p_sh[wv];

  const float inv_p2 = 1.f / (4.f * NN);
  for (int i = lane; i < 16 * 64; i += 32) {
    int pt = i >> 6, c = i & 63;
    int g = c >> 2;
    float s  = st_p2[(b * 16 + g) * 2 + 0];
    float q  = st_p2[(b * 16 + g) * 2 + 1];
    float mu = s * inv_p2;
    float var = q * inv_p2 - mu * mu;
    float rs = rsqrtf(var + EPSF);
    float v = (p2raw[(b * NN + nbase + pt) * 64 + c] - mu) * rs * g_p2[c] + bt_p2[c];
    psh[pt][c] = fmaxf(v, 0.f);
  }
  __syncthreads();

  v8f acc[8];
#pragma unroll
  for (int t = 0; t < 8; ++t) acc[t] = v8z();
#pragma unroll
  for (int kk = 0; kk < 64; kk += 4) {
    int k0 = kk + 2 * h;
    v2f bf;
    bf.x = psh[col][k0];
    bf.y = psh[col][k0 + 1];
#pragma unroll
    for (int t = 0; t < 8; ++t)
      acc[t] = wmma4(afrag(w_mlp2, 64, t * 16, kk, 64, lane), bf, acc[t]);
  }

  const float inv_sc = 1.f / (8.f * NN);
#pragma unroll
  for (int t = 0; t < 8; ++t)
#pragma unroll
    for (int r = 0; r < 8; ++r) {
      int o = t * 16 + r + 8 * h;
      int g = o >> 3;
      float s  = st_sc[(b * 16 + g) * 2 + 0];
      float q  = st_sc[(b * 16 + g) * 2 + 1];
      float mu = s * inv_sc;
      float var = q * inv_sc - mu * mu;
      float rs = rsqrtf(var + EPSF);
      int oi = (b * 128 + o) * NN + nbase + col;
      float scv = (scraw[oi] - mu) * rs * g_sc[o] + bt_sc[o];
      float v = acc[t][r] + b_mlp2[o] + scv;
      out[oi] = v > 0.f ? v : 0.01f * v;
    }
}

// ------------------------------------------------------------------
extern "C" void kernel_launch(void* const* d_in, const int* in_sizes, int n_in,
                              void* d_out, int out_size, void* d_ws,
                              size_t ws_size, hipStream_t stream) {
  (void)in_sizes; (void)n_in; (void)out_size; (void)ws_size;
  const float* coords  = (const float*)d_in[0];
  const float* feat    = (const float*)d_in[1];
  const int*   idx     = (const int*)d_in[2];
  const float* dist    = (const float*)d_in[3];
  const float* w_mlp1  = (const float*)d_in[4];
  const float* b_mlp1  = (const float*)d_in[5];
  const float* w_lse1  = (const float*)d_in[6];
  const float* b_lse1  = (const float*)d_in[7];
  const float* g_lse1  = (const float*)d_in[8];
  const float* bt_lse1 = (const float*)d_in[9];
  const float* w_pool1 = (const float*)d_in[10];
  const float* b_pool1 = (const float*)d_in[11];
  const float* g_pool1 = (const float*)d_in[12];
  const float* bt_pool1= (const float*)d_in[13];
  const float* w_lse2  = (const float*)d_in[14];
  const float* b_lse2  = (const float*)d_in[15];
  const float* g_lse2  = (const float*)d_in[16];
  const float* bt_lse2 = (const float*)d_in[17];
  const float* w_pool2 = (const float*)d_in[18];
  const float* b_pool2 = (const float*)d_in[19];
  const float* g_pool2 = (const float*)d_in[20];
  const float* bt_pool2= (const float*)d_in[21];
  const float* w_mlp2  = (const float*)d_in[22];
  const float* b_mlp2  = (const float*)d_in[23];
  const float* w_sc    = (const float*)d_in[24];
  const float* b_sc    = (const float*)d_in[25];
  const float* g_sc    = (const float*)d_in[26];
  const float* bt_sc   = (const float*)d_in[27];

  float* ws    = (float*)d_ws;
  float* x1    = ws + X1_OFF;
  float* x2    = ws + X2_OFF;
  float* scraw = ws + SC_OFF;
  float* p1raw = ws + P1_OFF;
  float* p2raw = ws + P2_OFF;
  float* st_sc = ws + ST_OFF;         // each slot: B*16*2 = 64 floats
  float* st_l1 = st_sc + 64;
  float* st_p1 = st_l1 + 64;
  float* st_l2 = st_p1 + 64;
  float* st_p2 = st_l2 + 64;

  hipMemsetAsync((void*)st_sc, 0, 5 * 64 * sizeof(float), stream);

  dim3 blk(256);
  int tile_blocks = (BB * NN / 16) / WAVES;   // 512
  int pt_blocks   = (BB * NN) / WAVES;        // 8192

  k_stage0<<<tile_blocks, blk, 0, stream>>>(feat, w_mlp1, b_mlp1, w_sc, b_sc,
                                            x1, scraw, st_sc);
  k_lse_stats<<<pt_blocks, blk, 0, stream>>>(coords, idx, dist, w_lse1, b_lse1,
                                             st_l1);
  k_lse_apply_pool<32><<<tile_blocks, blk, 0, stream>>>(
      coords, idx, dist, w_lse1, b_lse1, g_lse1, bt_lse1, st_l1,
      x1, w_pool1, b_pool1, p1raw, st_p1);
  k_gn_relu<<<(BB * NN * 32 + 255) / 256, blk, 0, stream>>>(
      p1raw, x2, st_p1, g_pool1, bt_pool1, 1.f / (2.f * NN), 32, BB * NN * 32);
  k_lse_stats<<<pt_blocks, blk, 0, stream>>>(coords, idx, dist, w_lse2, b_lse2,
                                             st_l2);
  k_lse_apply_pool<64><<<tile_blocks, blk, 0, stream>>>(
      coords, idx, dist, w_lse2, b_lse2, g_lse2, bt_lse2, st_l2,
      x2, w_pool2, b_pool2, p2raw, st_p2);
  k_final<<<tile_blocks, blk, 0, stream>>>(p2raw, st_p2, g_pool2, bt_pool2,
                                           w_mlp2, b_mlp2, scraw, st_sc, g_sc,
                                           bt_sc, (float*)d_out);
}